// FW_RNNCell_28509992911005
// MI455X (gfx1250) — compile-verified
//
#include <hip/hip_runtime.h>
#include <hip/hip_bf16.h>
#include <math.h>

#define B_   128
#define DIN_ 256
#define U_   512
#define EPS_ 0.001f

typedef __attribute__((ext_vector_type(2))) float v2f;
typedef __attribute__((ext_vector_type(8))) float v8f;

// -----------------------------------------------------------------------------
// Kernel 1: pre = prev_state @ W_h + inputs @ C   (fp32 WMMA 16x16x4)
//           h   = tanh(pre + bias)
// One 16x16 output tile per wave32. M tiles = 128/16 = 8, N tiles = 512/16 = 32
// -> 256 tiles = 256 waves. blockDim = 128 (4 waves), grid = 64 blocks.
//
// f32 A fragment (16x4, MxK): lanes 0-15 -> M=lane, VGPR0/1 = K,K+1;
//                             lanes 16-31 -> M=lane-16, VGPR0/1 = K+2,K+3.
// f32 B fragment (4x16, KxN): mirrored with N on lanes.
// f32 C/D (16x16): VGPR r -> M = r + 8*(lane>>4), N = lane&15.
// -----------------------------------------------------------------------------
__global__ __launch_bounds__(128) void fw_gemm_h_kernel(
    const float* __restrict__ prev,   // (128,512)
    const float* __restrict__ Wh,     // (512,512)
    const float* __restrict__ inp,    // (128,256)
    const float* __restrict__ Cm,     // (256,512)
    const float* __restrict__ bias,   // (512)
    float* __restrict__ pre,          // (128,512) ws
    float* __restrict__ h)            // (128,512) ws
{
    const int wave  = (blockIdx.x * blockDim.x + threadIdx.x) >> 5;
    const int lane  = threadIdx.x & 31;
    const int tileM = wave >> 5;       // 0..7
    const int tileN = wave & 31;       // 0..31
    const int half  = lane >> 4;       // selects K pair
    const int idx   = lane & 15;       // M index for A, N index for B

    const int m = tileM * 16 + idx;
    const int n = tileN * 16 + idx;

    v8f c = {};

    // prev_state (K = 512) x W_h
    for (int k = 0; k < U_; k += 4) {
        const int ka = k + 2 * half;
        v2f a, b;
        a.x = prev[m * U_ + ka];
        a.y = prev[m * U_ + ka + 1];
        b.x = Wh[(size_t)ka * U_ + n];
        b.y = Wh[(size_t)(ka + 1) * U_ + n];
        c = __builtin_amdgcn_wmma_f32_16x16x4_f32(
                false, a, false, b, (short)0, c, false, false);
    }
    // inputs (K = 256) x C
    for (int k = 0; k < DIN_; k += 4) {
        const int ka = k + 2 * half;
        v2f a, b;
        a.x = inp[m * DIN_ + ka];
        a.y = inp[m * DIN_ + ka + 1];
        b.x = Cm[(size_t)ka * U_ + n];
        b.y = Cm[(size_t)(ka + 1) * U_ + n];
        c = __builtin_amdgcn_wmma_f32_16x16x4_f32(
                false, a, false, b, (short)0, c, false, false);
    }

    const int gn = tileN * 16 + idx;
    const float bn = bias[gn];
#pragma unroll
    for (int r = 0; r < 8; ++r) {
        const int gm = tileM * 16 + r + 8 * half;
        const float p = c[r];
        pre[gm * U_ + gn] = p;
        h[gm * U_ + gn]   = tanhf(p + bn);
    }
}

// -----------------------------------------------------------------------------
// Kernel 2: A_new[b,i,j] = 0.9*A[b,i,j] + 0.5*h[b,i]*h[b,j]
// float4 per thread, fully coalesced streaming (the HBM-bound part).
// -----------------------------------------------------------------------------
__global__ __launch_bounds__(256) void fw_updateA_kernel(
    const float* __restrict__ A,      // (128,512,512)
    const float* __restrict__ h,      // (128,512)
    float* __restrict__ Anew)         // (128,512,512) -> d_out + 65536
{
    const size_t e4    = (size_t)blockIdx.x * blockDim.x + threadIdx.x;
    const int per_b    = U_ * U_ / 4;           // 65536 float4 per batch
    const int b        = (int)(e4 / per_b);
    const int rem      = (int)(e4 % per_b);
    const int i        = rem / (U_ / 4);        // row
    const int j4       = (rem % (U_ / 4)) * 4;  // col start

    const float  hi = h[b * U_ + i];
    const float4 hj = *(const float4*)(h + b * U_ + j4);
    const size_t off = ((size_t)b * U_ + i) * U_ + j4;
    const float4 av = *(const float4*)(A + off);

    float4 o;
    o.x = fmaf(0.5f * hi, hj.x, 0.9f * av.x);
    o.y = fmaf(0.5f * hi, hj.y, 0.9f * av.y);
    o.z = fmaf(0.5f * hi, hj.z, 0.9f * av.z);
    o.w = fmaf(0.5f * hi, hj.w, 0.9f * av.w);
    *(float4*)(Anew + off) = o;
}

// -----------------------------------------------------------------------------
// Kernel 3: 3 recurrent steps. One block per batch, thread v owns column v.
//   x = tanh(pre[b,v] + sum_u h_s[u] * A_new[b,u,v])   (coalesced over v)
//   h_s = LayerNorm(x) * gamma + beta
// A_new re-reads should hit L2 (128 MB < 192 MB).
// -----------------------------------------------------------------------------
__global__ __launch_bounds__(512) void fw_recur_kernel(
    const float* __restrict__ pre,    // (128,512) ws
    const float* __restrict__ h,      // (128,512) ws
    const float* __restrict__ Anew,   // (128,512,512)
    const float* __restrict__ gamma,  // (512)
    const float* __restrict__ beta,   // (512)
    float* __restrict__ out)          // (128,512) -> d_out
{
    const int b = blockIdx.x;
    const int v = threadIdx.x;

    __shared__ float hs[U_];
    __shared__ float rs[16], rq[16];
    __shared__ float mean_s, rstd_s;

    hs[v] = h[b * U_ + v];
    const float pv = pre[b * U_ + v];
    const float gv = gamma[v];
    const float bv = beta[v];
    const float* __restrict__ Ab = Anew + (size_t)b * U_ * U_;
    __syncthreads();

    for (int step = 0; step < 3; ++step) {
        float acc = 0.0f;
#pragma unroll 8
        for (int u = 0; u < U_; ++u)
            acc = fmaf(hs[u], Ab[(size_t)u * U_ + v], acc);

        const float x = tanhf(pv + acc);

        // block-wide LayerNorm reduction (wave32 shuffle + LDS)
        float s = x, q = x * x;
        for (int off = 16; off > 0; off >>= 1) {
            s += __shfl_down(s, off, 32);
            q += __shfl_down(q, off, 32);
        }
        const int wid = v >> 5, ln = v & 31;
        if (ln == 0) { rs[wid] = s; rq[wid] = q; }
        __syncthreads();
        if (v == 0) {
            float ts = 0.0f, tq = 0.0f;
#pragma unroll
            for (int i = 0; i < 16; ++i) { ts += rs[i]; tq += rq[i]; }
            const float m  = ts * (1.0f / U_);
            const float vr = tq * (1.0f / U_) - m * m;
            mean_s = m;
            rstd_s = rsqrtf(vr + EPS_);
        }
        __syncthreads();

        hs[v] = (x - mean_s) * rstd_s * gv + bv;
        __syncthreads();
    }

    out[b * U_ + v] = hs[v];
}

extern "C" void kernel_launch(void* const* d_in, const int* in_sizes, int n_in,
                              void* d_out, int out_size, void* d_ws, size_t ws_size,
                              hipStream_t stream) {
    const float* inputs = (const float*)d_in[0];  // (128,256)
    const float* prev   = (const float*)d_in[1];  // (128,512)
    const float* A      = (const float*)d_in[2];  // (128,512,512)
    const float* Cm     = (const float*)d_in[3];  // (256,512)
    const float* Wh     = (const float*)d_in[4];  // (512,512)
    const float* bias   = (const float*)d_in[5];  // (512)
    const float* gamma  = (const float*)d_in[6];  // (512)
    const float* beta   = (const float*)d_in[7];  // (512)

    float* out_h  = (float*)d_out;                 // first 128*512 floats
    float* out_A  = (float*)d_out + (size_t)B_ * U_;  // 128*512*512 floats

    float* pre = (float*)d_ws;                     // 128*512
    float* h   = pre + (size_t)B_ * U_;            // 128*512

    // K1: fused GEMM + tanh (WMMA fp32). 256 tiles / 4 waves per block.
    fw_gemm_h_kernel<<<64, 128, 0, stream>>>(prev, Wh, inputs, Cm, bias, pre, h);

    // K2: streaming fast-weight update -> d_out A region.
    const int n4 = B_ * U_ * U_ / 4;               // 8388608 float4
    fw_updateA_kernel<<<n4 / 256, 256, 0, stream>>>(A, h, out_A);

    // K3: 3 recurrent steps + LayerNorm, one block per batch.
    fw_recur_kernel<<<B_, U_, 0, stream>>>(pre, h, out_A, gamma, beta, out_h);
}